// MaxROI_38534446579959
// MI455X (gfx1250) — compile-verified
//
#include <hip/hip_runtime.h>

namespace {
constexpr int BATCH   = 256;
constexpr int NBOX    = 65536;
constexpr int KNMS    = 24;           // K
constexpr int MAXN    = 5;            // MAX_NUM
constexpr int KSEL    = KNMS + MAXN;  // 29 candidates
constexpr int THREADS = 128;          // 4 wave32 waves
constexpr int WAVES   = THREADS / 32;
constexpr int DEPTH   = 16;           // async ring depth
constexpr int EPC     = THREADS * 2;  // elements per chunk (b128 = 2 score pairs/lane)
constexpr int CHUNKS  = NBOX / EPC;   // 256 chunks
constexpr float IOU_THRESH = 0.5f;
constexpr float BIGF = 1e30f;
}

// ---- CDNA5 async global->LDS streaming (ASYNCcnt, §15.18 op 98), NT hint ----
__device__ __forceinline__ void async_load_b128(const void* gp, unsigned lds_off) {
  asm volatile("global_load_async_to_lds_b128 %0, %1, off th:TH_LOAD_NT"
               :: "v"(lds_off), "v"((unsigned long long)(size_t)gp)
               : "memory");
}
__device__ __forceinline__ void wait_async_le15() {
  asm volatile("s_wait_asynccnt 15" ::: "memory");
}
__device__ __forceinline__ void wait_async_le0() {
  asm volatile("s_wait_asynccnt 0" ::: "memory");
}

__device__ __forceinline__ bool better(float av, int ai, float bv, int bi) {
  // jax.lax.top_k order: value descending, ties -> smaller index first
  return (av > bv) || (av == bv && ai < bi);
}

__global__ __launch_bounds__(THREADS, 1)
void maxroi_topk_nms_kernel(const float* __restrict__ boxes,
                            const float* __restrict__ scores,
                            float* __restrict__ out) {
  __shared__ float          lv[THREADS * KSEL];    // per-thread top-29 values (14.5 KB)
  __shared__ unsigned short li[THREADS * KSEL];    // per-thread top-29 indices (7.25 KB)
  __shared__ float4         stage[DEPTH][THREADS]; // async ring (32 KB)
  __shared__ float  rv[WAVES];
  __shared__ int    ri[WAVES];
  __shared__ int    rs[WAVES];
  __shared__ float  selv[KSEL];
  __shared__ int    seli[KSEL];
  __shared__ float4 sbox[KSEL];

  const int tid = threadIdx.x;
  const int b   = blockIdx.x;
  // batch base, viewed as float4 = two (s0,s1) score pairs
  const float4* __restrict__ sc4 =
      reinterpret_cast<const float4*>(scores) + (size_t)b * (NBOX / 2);

  float*          mylv = &lv[tid * KSEL];
  unsigned short* myli = &li[tid * KSEL];

  // ---------------- streaming scan: exact per-thread top-29 ----------------
  int   cnt  = 0;
  float minv = BIGF;
  int   minp = 0;

  auto push = [&](float d, int idx) {
    if (cnt < KSEL) {
      mylv[cnt] = d; myli[cnt] = (unsigned short)idx;
      if (++cnt == KSEL) {
        minv = mylv[0]; minp = 0;
#pragma unroll
        for (int j = 1; j < KSEL; ++j) {
          float x = mylv[j];
          if (x < minv) { minv = x; minp = j; }
        }
      }
    } else if (d > minv) {                 // strict >: tie keeps smaller index
      mylv[minp] = d; myli[minp] = (unsigned short)idx;
      minv = mylv[0]; minp = 0;
#pragma unroll
      for (int j = 1; j < KSEL; ++j) {
        float x = mylv[j];
        if (x < minv) { minv = x; minp = j; }
      }
    }
  };

  // prime DEPTH-1 ring slots (slot DEPTH-1 stays free for the first in-loop issue)
#pragma unroll
  for (int p = 0; p < DEPTH - 1; ++p) {
    async_load_b128(sc4 + (size_t)p * THREADS + tid,
                    (unsigned)(size_t)&stage[p][tid]);
  }

  // steady state: iteration `it` refills the slot consumed at iteration `it-1`
  // (slot (it-1)&15 hosts chunk it+DEPTH-1); DScnt ordering is implied by the
  // previous iteration's consumption preceding this asm in program order.
  for (int it = 0; it <= CHUNKS - DEPTH; ++it) {
    const int c = it + DEPTH - 1;               // chunk to issue now
    async_load_b128(sc4 + (size_t)c * THREADS + tid,
                    (unsigned)(size_t)&stage[c & (DEPTH - 1)][tid]);
    wait_async_le15();                          // oldest chunk (it) has landed
    const float4 s = stage[it & (DEPTH - 1)][tid];
    const int i0 = it * EPC + 2 * tid;
    push(s.y - s.x, i0);                        // monotonic surrogate for softmax[:,1]
    push(s.w - s.z, i0 + 1);
  }
  wait_async_le0();                             // drain tail (last DEPTH-1 chunks)
  for (int it = CHUNKS - DEPTH + 1; it < CHUNKS; ++it) {
    const float4 s = stage[it & (DEPTH - 1)][tid];
    const int i0 = it * EPC + 2 * tid;
    push(s.y - s.x, i0);
    push(s.w - s.z, i0 + 1);
  }
  __syncthreads();

  // ------------- merge: exact sorted top-29 over 128*29 candidates -------------
  for (int r = 0; r < KSEL; ++r) {
    float bv = mylv[0]; int bi = myli[0]; int bs = tid * KSEL;
#pragma unroll
    for (int j = 1; j < KSEL; ++j) {
      float v = mylv[j]; int ix = myli[j];
      if (better(v, ix, bv, bi)) { bv = v; bi = ix; bs = tid * KSEL + j; }
    }
#pragma unroll
    for (int off = 16; off > 0; off >>= 1) {    // wave32 reduce
      float v  = __shfl_down(bv, off, 32);
      int   ix = __shfl_down(bi, off, 32);
      int   sl = __shfl_down(bs, off, 32);
      if (better(v, ix, bv, bi)) { bv = v; bi = ix; bs = sl; }
    }
    if ((tid & 31) == 0) { int w = tid >> 5; rv[w] = bv; ri[w] = bi; rs[w] = bs; }
    __syncthreads();
    if (tid == 0) {
      float v = rv[0]; int ix = ri[0]; int sl = rs[0];
      for (int w = 1; w < WAVES; ++w)
        if (better(rv[w], ri[w], v, ix)) { v = rv[w]; ix = ri[w]; sl = rs[w]; }
      selv[r] = v; seli[r] = ix;
      lv[sl] = -BIGF; li[sl] = 0xFFFFu;         // retire winner
    }
    __syncthreads();
  }

  // ---------------- gather the 29 candidate boxes ----------------
  if (tid < KSEL) {
    sbox[tid] = reinterpret_cast<const float4*>(boxes)[(size_t)b * NBOX + seli[tid]];
  }
  __syncthreads();

  // ---------------- NMS clustering on wave 0 (wave32 lane-parallel) ----------------
  if (tid < 32) {
    const int lane = tid;
    bool mask = lane < KNMS;
    float4 bx = sbox[lane < KSEL ? lane : 0];
    float x1 = bx.x, y1 = bx.y, x2 = bx.z, y2 = bx.w;
    float sc = (lane < KSEL) ? selv[lane] : -BIGF;
    float* outp = out + (size_t)b * (MAXN * 4);

    for (int j = 0; j < MAXN - 1; ++j) {
      // argmax over masked scores (tie -> lowest position)
      float mv = mask ? sc : -BIGF;
      int   ml = lane;
#pragma unroll
      for (int off = 16; off > 0; off >>= 1) {
        float v = __shfl_down(mv, off, 32);
        int   l = __shfl_down(ml, off, 32);
        if (v > mv || (v == mv && l < ml)) { mv = v; ml = l; }
      }
      ml = __shfl(ml, 0, 32);
      const float ax1 = __shfl(x1, ml, 32), ay1 = __shfl(y1, ml, 32);
      const float ax2 = __shfl(x2, ml, 32), ay2 = __shfl(y2, ml, 32);

      const float ix1 = fmaxf(ax1, x1), iy1 = fmaxf(ay1, y1);
      const float ix2 = fminf(ax2, x2), iy2 = fminf(ay2, y2);
      const float inter = fmaxf(ix2 - ix1, 0.f) * fmaxf(iy2 - iy1, 0.f);
      const float areaA = (ax2 - ax1) * (ay2 - ay1);
      const float areaB = (x2 - x1) * (y2 - y1);
      const float iou   = inter / (areaA + areaB - inter);

      const bool over = mask && (iou >= IOU_THRESH);
      float cx1 = over ? x1 :  BIGF, cy1 = over ? y1 :  BIGF;
      float cx2 = over ? x2 : -BIGF, cy2 = over ? y2 : -BIGF;
#pragma unroll
      for (int off = 16; off > 0; off >>= 1) {
        cx1 = fminf(cx1, __shfl_down(cx1, off, 32));
        cy1 = fminf(cy1, __shfl_down(cy1, off, 32));
        cx2 = fmaxf(cx2, __shfl_down(cx2, off, 32));
        cy2 = fmaxf(cy2, __shfl_down(cy2, off, 32));
      }
      if (lane == 0) {
        outp[j * 4 + 0] = cx1; outp[j * 4 + 1] = cy1;
        outp[j * 4 + 2] = cx2; outp[j * 4 + 3] = cy2;
      }

      const bool nxt = mask && (iou < IOU_THRESH);
      const bool empty = (__ballot(nxt) == 0ull);
      if (empty) {
        mask = (lane == 0);
        if (lane == 0) {                 // placeholder promotion
          const float4 pb = sbox[KNMS + j];
          x1 = pb.x; y1 = pb.y; x2 = pb.z; y2 = pb.w;
          sc = selv[KNMS + j];
        }
      } else {
        mask = nxt;
      }
    }
    if (lane == 0) {                     // rois.append(top_boxes[K + MAX_NUM - 2])
      const float4 lb = sbox[KNMS + MAXN - 2];
      outp[16] = lb.x; outp[17] = lb.y; outp[18] = lb.z; outp[19] = lb.w;
    }
  }
}

extern "C" void kernel_launch(void* const* d_in, const int* in_sizes, int n_in,
                              void* d_out, int out_size, void* d_ws, size_t ws_size,
                              hipStream_t stream) {
  (void)in_sizes; (void)n_in; (void)out_size; (void)d_ws; (void)ws_size;
  const float* boxes  = (const float*)d_in[0];   // (B, N, 4) f32
  const float* scores = (const float*)d_in[1];   // (B, N, 2) f32
  float* out = (float*)d_out;                    // (B, 5, 4) f32
  maxroi_topk_nms_kernel<<<dim3(BATCH), dim3(THREADS), 0, stream>>>(boxes, scores, out);
}